// Model_39676907886571
// MI455X (gfx1250) — compile-verified
//
#include <hip/hip_runtime.h>
#include <math.h>

typedef float v2f __attribute__((ext_vector_type(2)));
typedef float v8f __attribute__((ext_vector_type(8)));

constexpr int HN = 8;      // heads (== batch)
constexpr int SN = 2048;   // sequence
constexpr int DN = 3;      // head projections

// LDS layout (float indices). x1 region is reused as per-wave P staging
// once Q fragments have been built.
constexpr int X1_OFF = 0;                    // 2048*3 = 6144 floats
constexpr int K_OFF  = 6144;                 // 2048*4 (col 3 zero-padded)
constexpr int VT_STR = 2052;                 // 2048 + 4 pad: bank-conflict-free
constexpr int VT_OFF = 6144 + 8192;          // V^T: 5 x 2052 (row 4 = zeros)
constexpr int SMEM_FLOATS = 6144 + 8192 + 5 * VT_STR;   // ~96 KB (<< 320 KB/WGP)

// log2(e) / sqrt(3): fold the 1/sqrt(i0) scale into the exp2-domain softmax.
#define SOFTMAX_SCALE 0.8329862557658682f

// Raw maxnum without the compiler's canonicalize-before-maxnum expansion
// (inputs here are products/sums of finite data; sNaN semantics irrelevant).
__device__ __forceinline__ float fmax_raw(float a, float b) {
    float d;
    asm("v_max_num_f32_e32 %0, %1, %2" : "=v"(d) : "v"(a), "v"(b));
    return d;
}

// Butterfly max over 16-lane rows. DPP rides src0 of v_max_num_f32 directly
// (one instruction per step, no mov/canonicalize): xor1, xor2 via quad_perm;
// then (values quad-uniform) row_half_mirror == xor4, row_mirror == xor8.
__device__ __forceinline__ float rowmax16(float x) {
    float t;
    asm("v_max_num_f32_dpp %0, %1, %1 quad_perm:[1,0,3,2] row_mask:0xf bank_mask:0xf bound_ctrl:1"
        : "=v"(t) : "v"(x));
    asm("v_max_num_f32_dpp %0, %1, %1 quad_perm:[2,3,0,1] row_mask:0xf bank_mask:0xf bound_ctrl:1"
        : "=v"(x) : "v"(t));
    asm("v_max_num_f32_dpp %0, %1, %1 row_half_mirror row_mask:0xf bank_mask:0xf bound_ctrl:1"
        : "=v"(t) : "v"(x));
    asm("v_max_num_f32_dpp %0, %1, %1 row_mirror row_mask:0xf bank_mask:0xf bound_ctrl:1"
        : "=v"(x) : "v"(t));
    return x;
}

__global__ __launch_bounds__(128)
void attn_view_kernel(const float* __restrict__ x1,
                      const float* __restrict__ qw_g,
                      const float* __restrict__ kw_g,
                      const float* __restrict__ vw_g,
                      float* __restrict__ out)
{
    extern __shared__ float smem[];
    const int tid  = threadIdx.x;
    const int lane = tid & 31;
    const int wave = tid >> 5;
    const int h    = blockIdx.x >> 5;          // 32 blocks per head
    const int sblk = blockIdx.x & 31;          // 64 rows per block (16/wave)
    const int s_tile_base = sblk * 64 + wave * 16;

    const float* x1h = x1 + h * (SN * DN);

    // ---- per-head 3x3 weights (uniform addresses -> scalar loads) ----
    float qw[9], kw[9], vw[9];
    #pragma unroll
    for (int i = 0; i < 9; ++i) {
        qw[i] = qw_g[h * 9 + i];
        kw[i] = kw_g[h * 9 + i];
        vw[i] = vw_g[h * 9 + i];
    }

    // ---- phase 1: stage x1[h] (2048x3 f32 = 24KB) into LDS ----
    {
        const float4* src = (const float4*)x1h;
        float4* dst = (float4*)&smem[X1_OFF];
        #pragma unroll
        for (int i = 0; i < 12; ++i)           // 6144 floats / 4 / 128 thr
            dst[tid + i * 128] = src[tid + i * 128];
    }
    __syncthreads();

    // ---- phase 2: materialize K (2048x4, col3=0) and V^T (5x2052) honoring
    //      the reshape-VIEW: element (t,p) comes from flat index f = 3t+p of
    //      the (3,2048) matmul result: row f>>11 of W, column f&2047 of x1.
    //      V^T row 3 = 1.0 (PV WMMA accumulates the softmax denominator in
    //      output column 3 for free); V^T row 4 = 0 (zero source for lanes
    //      whose B column is 4..15 -> masking by addressing, not VALU).
    for (int e = tid; e < SN * 4; e += 128) {
        int t = e >> 2, pp = e & 3;
        float kv = 0.f, vv = 1.f;
        if (pp < 3) {
            int f  = t * 3 + pp;
            int i  = f >> 11;
            int tt = f & 2047;
            const float* xr = &smem[X1_OFF + tt * 3];
            kv = kw[i*3+0]*xr[0] + kw[i*3+1]*xr[1] + kw[i*3+2]*xr[2];
            vv = vw[i*3+0]*xr[0] + vw[i*3+1]*xr[1] + vw[i*3+2]*xr[2];
        }
        smem[K_OFF + e] = kv;
        smem[VT_OFF + pp * VT_STR + t] = vv;
    }
    for (int e = tid; e < VT_STR; e += 128)    // zero row 4
        smem[VT_OFF + 4 * VT_STR + e] = 0.f;

    // ---- Q fragment in WMMA A-layout (16x4 f32): lane l holds row m=l&15,
    //      K-slots k0,k0+1 with k0 = (l>>4)*2; slot 3 is zero pad.
    //      The softmax scale is folded into Q so qk comes out pre-scaled. ----
    const int m   = lane & 15;
    const int k0  = (lane >> 4) * 2;
    const int hi8 = (lane >> 4) * 8;           // C/D rows: r (low) / r+8 (high)
    v2f aq;
    #pragma unroll
    for (int j = 0; j < 2; ++j) {
        int kk = k0 + j;
        float q = 0.f;
        if (kk < 3) {
            int f  = (s_tile_base + m) * 3 + kk;
            int i  = f >> 11;
            int tt = f & 2047;
            const float* xr = &smem[X1_OFF + tt * 3];
            q = (qw[i*3+0]*xr[0] + qw[i*3+1]*xr[1] + qw[i*3+2]*xr[2]) * SOFTMAX_SCALE;
        }
        aq[j] = q;
    }
    __syncthreads();   // K/V ready; x1 region now free -> P staging

    float rmax[8];
    v8f acc = {};
    #pragma unroll
    for (int r = 0; r < 8; ++r) rmax[r] = -__builtin_inff();

    float* pstage = &smem[X1_OFF + wave * 256];   // 16x16 P tile per wave
    // Per-lane V^T base: columns n>=4 read the zero row (no masking VALU).
    const float* vbase = &smem[VT_OFF + ((m < 4) ? m : 4) * VT_STR];

    // No barriers in this loop: P staging is per-wave private and same-wave
    // LDS instructions are hardware in-order (DScnt).
    for (int tb = 0; tb < SN / 16; ++tb) {
        const int tbase = tb * 16;

        // B fragment of K^T (4x16): lane = column n=m, contiguous K pair.
        v2f bk = *(const v2f*)&smem[K_OFF + (tbase + m) * 4 + k0];
        v8f zero = {};
        v8f qk = __builtin_amdgcn_wmma_f32_16x16x4_f32(
            false, aq, false, bk, (short)0, zero, false, false);

        // ---- online softmax (exp2 domain; qk is pre-scaled). qk VGPR r
        //      holds rows r / r+8 in low/high halves; DPP rowmax matches. ----
        #pragma unroll
        for (int r = 0; r < 8; ++r) {
            float x  = qk[r];
            float nm = fmax_raw(rmax[r], rowmax16(x));
            float pv = __builtin_amdgcn_exp2f(x - nm);
            float cr = __builtin_amdgcn_exp2f(rmax[r] - nm);
            rmax[r]  = nm;
            acc[r]  *= cr;                         // rescale before PV adds
            pstage[(r + hi8) * 16 + m] = pv;       // C-layout -> LDS
        }

        // ---- PV: (16x16 P) x (16x4 [V | 1]) as 4 chained K=4 WMMAs.
        //      Column 3 of acc accumulates the softmax denominator. ----
        #pragma unroll
        for (int c = 0; c < 4; ++c) {
            // A chunk: P[:, 4c+k0 .. 4c+k0+1], re-read in A-layout
            v2f ap = *(const v2f*)&pstage[m * 16 + 4 * c + k0];
            // B chunk: V^T[n, t0..t0+1] -> one b64 load (zero row for n>=4)
            const int t0 = tbase + 4 * c + k0;
            v2f bv = *(const v2f*)&vbase[t0];
            acc = __builtin_amdgcn_wmma_f32_16x16x4_f32(
                false, ap, false, bv, (short)0, acc, false, false);
        }
    }

    // ---- epilogue: denominator lives in column 3 of each acc row ----
    #pragma unroll
    for (int r = 0; r < 8; ++r) {
        float denom = __shfl(acc[r], 3, 16);       // lane col 3 of this row
        if (m < 3) {
            int s = s_tile_base + r + hi8;
            out[(h * 3 + m) * SN + s] = acc[r] / denom;
        }
    }
}

extern "C" void kernel_launch(void* const* d_in, const int* in_sizes, int n_in,
                              void* d_out, int out_size, void* d_ws, size_t ws_size,
                              hipStream_t stream) {
    const float* x1 = (const float*)d_in[0];
    const float* qw = (const float*)d_in[1];
    const float* kw = (const float*)d_in[2];
    const float* vw = (const float*)d_in[3];
    float* out = (float*)d_out;
    (void)in_sizes; (void)n_in; (void)out_size; (void)d_ws; (void)ws_size;

    dim3 grid(HN * 32);   // 8 heads x 32 s-blocks (64 rows each)
    dim3 block(128);      // 4 waves; one 16-row s-tile per wave
    size_t smem = SMEM_FLOATS * sizeof(float);   // ~96 KB dynamic LDS
    attn_view_kernel<<<grid, block, smem, stream>>>(x1, qw, kw, vw, out);
}